// GraphMixerEncoder_72292889526686
// MI455X (gfx1250) — compile-verified
//
#include <hip/hip_runtime.h>
#include <math.h>

// ---------------------------------------------------------------------------
// GraphMixer encoder forward, fp32 end-to-end on gfx1250.
// Matmuls use V_WMMA_F32_16X16X4_F32 (exact fp32 matrix path on CDNA5).
// M=32 register blocking: each B fragment (global loads) feeds 2 WMMAs.
// ---------------------------------------------------------------------------

typedef float v2f __attribute__((ext_vector_type(2)));
typedef float v8f __attribute__((ext_vector_type(8)));

#define BB 6144
#define TT 32
#define CC 128
#define TDIM 100
#define NDIM 128
#define EGN (BB * 64)
#define THID 16
#define CHID 512
#define KIN (CC + TDIM) // 228

// D = A(16x4 f32) * B(4x16 f32) + C(16x16 f32), wave32, one matrix per wave.
__device__ __forceinline__ v8f wmma4(v2f a, v2f b, v8f c) {
  return __builtin_amdgcn_wmma_f32_16x16x4_f32(
      /*neg_a=*/false, a, /*neg_b=*/false, b,
      /*c_mod=*/(short)0, c, /*reuse_a=*/false, /*reuse_b=*/false);
}

__device__ __forceinline__ float gelu_exact(float x) {
  return 0.5f * x * (1.0f + erff(x * 0.70710678118654752f));
}

// ---------------------------------------------------------------------------
// Kernel 1: z[B*T,128] = concat(nbr_feats, cos(dt*w+b)) @ proj_w + proj_b
// One block = 32-row M tile (2 sub-tiles); 8 waves = 8 N-tiles of 16.
// ---------------------------------------------------------------------------
__global__ __launch_bounds__(256) void proj_kernel(
    const float* __restrict__ nbr_feats, const float* __restrict__ times,
    const float* __restrict__ nbr_times, const float* __restrict__ time_w,
    const float* __restrict__ time_b, const float* __restrict__ proj_w,
    const float* __restrict__ proj_b, float* __restrict__ z) {
  __shared__ float at[32 * 236]; // 32 x 228, padded stride (gcd(236,64)=4)
  const int tid = threadIdx.x;
  const int m0 = blockIdx.x * 32; // row base in [B*T]

  for (int idx = tid; idx < 32 * KIN; idx += 256) {
    int r = idx / KIN, c = idx - r * KIN;
    int gr = m0 + r;
    float v;
    if (c < CC) {
      v = nbr_feats[gr * CC + c];
    } else {
      int k = c - CC;
      float dt = times[gr / TT] - nbr_times[gr];
      v = cosf(dt * time_w[k] + time_b[k]);
    }
    at[r * 236 + c] = v;
  }
  __syncthreads();

  const int lane = tid & 31, wave = tid >> 5;
  const int half = lane >> 4, l16 = lane & 15;
  const int n0 = wave * 16;

  v8f acc0, acc1;
  {
    float bias = proj_b[n0 + l16];
#pragma unroll
    for (int i = 0; i < 8; i++) { acc0[i] = bias; acc1[i] = bias; }
  }
  for (int k0 = 0; k0 < KIN; k0 += 4) {
    const int kk = k0 + 2 * half;
    v2f bb, a0, a1;
    bb.x = proj_w[kk * CC + n0 + l16];
    bb.y = proj_w[(kk + 1) * CC + n0 + l16];
    a0.x = at[l16 * 236 + kk];
    a0.y = at[l16 * 236 + kk + 1];
    a1.x = at[(l16 + 16) * 236 + kk];
    a1.y = at[(l16 + 16) * 236 + kk + 1];
    acc0 = wmma4(a0, bb, acc0);
    acc1 = wmma4(a1, bb, acc1);
  }
#pragma unroll
  for (int i = 0; i < 8; i++) {
    z[(m0 + i + 8 * half) * CC + n0 + l16] = acc0[i];
    z[(m0 + 16 + i + 8 * half) * CC + n0 + l16] = acc1[i];
  }
}

// ---------------------------------------------------------------------------
// Kernel 2: token mixing for layer l.
// z[b] (32x128) in LDS; thread c does LN over T + FF(32->16->32) in registers.
// ---------------------------------------------------------------------------
__global__ __launch_bounds__(128) void token_kernel(
    const float* __restrict__ tok_gamma, const float* __restrict__ tok_beta,
    const float* __restrict__ tok_w1, const float* __restrict__ tok_b1,
    const float* __restrict__ tok_w2, const float* __restrict__ tok_b2,
    float* __restrict__ z, int layer) {
  __shared__ float zt[TT * CC];
  const int b = blockIdx.x, tid = threadIdx.x;
  for (int idx = tid; idx < TT * CC; idx += 128)
    zt[idx] = z[b * TT * CC + idx];
  __syncthreads();

  const int c = tid;
  float x[TT];
  float mu = 0.f;
#pragma unroll
  for (int t = 0; t < TT; t++) {
    x[t] = zt[t * CC + c];
    mu += x[t];
  }
  mu *= (1.0f / TT);
  float var = 0.f;
#pragma unroll
  for (int t = 0; t < TT; t++) {
    float d = x[t] - mu;
    var += d * d;
  }
  var *= (1.0f / TT);
  float rs = rsqrtf(var + 1e-5f);

  const float* ga = tok_gamma + layer * TT;
  const float* be = tok_beta + layer * TT;
  float hn[TT];
#pragma unroll
  for (int t = 0; t < TT; t++) hn[t] = (x[t] - mu) * rs * ga[t] + be[t];

  const float* w1 = tok_w1 + layer * TT * THID;
  const float* b1 = tok_b1 + layer * THID;
  const float* w2 = tok_w2 + layer * THID * TT;
  const float* b2 = tok_b2 + layer * TT;

  float g[THID];
#pragma unroll
  for (int j = 0; j < THID; j++) {
    float s = b1[j];
#pragma unroll
    for (int t = 0; t < TT; t++) s += hn[t] * w1[t * THID + j];
    g[j] = gelu_exact(s);
  }
#pragma unroll
  for (int t = 0; t < TT; t++) {
    float s = b2[t];
#pragma unroll
    for (int j = 0; j < THID; j++) s += g[j] * w2[j * TT + t];
    z[(b * TT + t) * CC + c] = x[t] + s;
  }
}

// ---------------------------------------------------------------------------
// Kernel 3: channel mixing for layer l (fused LN -> GEMM1 -> GELU -> GEMM2 ->
// residual). One block = 32-row M tile (2 sub-tiles); 8 waves.
// ---------------------------------------------------------------------------
__global__ __launch_bounds__(256) void channel_kernel(
    const float* __restrict__ ch_gamma, const float* __restrict__ ch_beta,
    const float* __restrict__ ch_w1, const float* __restrict__ ch_b1,
    const float* __restrict__ ch_w2, const float* __restrict__ ch_b2,
    float* __restrict__ z, int layer) {
  __shared__ float zt[32 * 132];   // raw rows (residual), padded
  __shared__ float hn[32 * 132];   // layernormed rows, padded
  __shared__ float gact[32 * 516]; // gelu(h@w1+b1), 32x512 padded
  __shared__ float mu_s[32], rs_s[32];
  const int tid = threadIdx.x;
  const int m0 = blockIdx.x * 32;

  for (int idx = tid; idx < 32 * CC; idx += 256) {
    int r = idx >> 7, c = idx & 127;
    zt[r * 132 + c] = z[(m0 + r) * CC + c];
  }
  __syncthreads();

  if (tid < 32) {
    float mu = 0.f;
    for (int c2 = 0; c2 < CC; c2++) mu += zt[tid * 132 + c2];
    mu *= (1.0f / CC);
    float var = 0.f;
    for (int c2 = 0; c2 < CC; c2++) {
      float d = zt[tid * 132 + c2] - mu;
      var += d * d;
    }
    var *= (1.0f / CC);
    mu_s[tid] = mu;
    rs_s[tid] = rsqrtf(var + 1e-5f);
  }
  __syncthreads();

  const float* gam = ch_gamma + layer * CC;
  const float* bet = ch_beta + layer * CC;
  for (int idx = tid; idx < 32 * CC; idx += 256) {
    int r = idx >> 7, c = idx & 127;
    hn[r * 132 + c] = (zt[r * 132 + c] - mu_s[r]) * rs_s[r] * gam[c] + bet[c];
  }
  __syncthreads();

  const int lane = tid & 31, wave = tid >> 5;
  const int half = lane >> 4, l16 = lane & 15;

  // GEMM1: [32x128] @ [128x512] -> gelu -> gact
  const float* w1 = ch_w1 + layer * CC * CHID;
  const float* b1 = ch_b1 + layer * CHID;
  for (int nt = 0; nt < 4; nt++) {
    const int n0 = (wave * 4 + nt) * 16;
    v8f acc0, acc1;
    {
      float bias = b1[n0 + l16];
#pragma unroll
      for (int i = 0; i < 8; i++) { acc0[i] = bias; acc1[i] = bias; }
    }
    for (int k0 = 0; k0 < CC; k0 += 4) {
      const int kk = k0 + 2 * half;
      v2f bb, a0, a1;
      bb.x = w1[kk * CHID + n0 + l16];
      bb.y = w1[(kk + 1) * CHID + n0 + l16];
      a0.x = hn[l16 * 132 + kk];
      a0.y = hn[l16 * 132 + kk + 1];
      a1.x = hn[(l16 + 16) * 132 + kk];
      a1.y = hn[(l16 + 16) * 132 + kk + 1];
      acc0 = wmma4(a0, bb, acc0);
      acc1 = wmma4(a1, bb, acc1);
    }
#pragma unroll
    for (int i = 0; i < 8; i++) {
      gact[(i + 8 * half) * 516 + n0 + l16] = gelu_exact(acc0[i]);
      gact[(16 + i + 8 * half) * 516 + n0 + l16] = gelu_exact(acc1[i]);
    }
  }
  __syncthreads();

  // GEMM2: [32x512] @ [512x128] + residual
  const float* w2 = ch_w2 + layer * CHID * CC;
  const float* b2 = ch_b2 + layer * CC;
  {
    const int n0 = wave * 16;
    v8f acc0, acc1;
    {
      float bias = b2[n0 + l16];
#pragma unroll
      for (int i = 0; i < 8; i++) { acc0[i] = bias; acc1[i] = bias; }
    }
    for (int k0 = 0; k0 < CHID; k0 += 4) {
      const int kk = k0 + 2 * half;
      v2f bb, a0, a1;
      bb.x = w2[kk * CC + n0 + l16];
      bb.y = w2[(kk + 1) * CC + n0 + l16];
      a0.x = gact[l16 * 516 + kk];
      a0.y = gact[l16 * 516 + kk + 1];
      a1.x = gact[(l16 + 16) * 516 + kk];
      a1.y = gact[(l16 + 16) * 516 + kk + 1];
      acc0 = wmma4(a0, bb, acc0);
      acc1 = wmma4(a1, bb, acc1);
    }
#pragma unroll
    for (int i = 0; i < 8; i++) {
      int r0 = i + 8 * half;
      int r1 = 16 + i + 8 * half;
      z[(m0 + r0) * CC + n0 + l16] = zt[r0 * 132 + n0 + l16] + acc0[i];
      z[(m0 + r1) * CC + n0 + l16] = zt[r1 * 132 + n0 + l16] + acc1[i];
    }
  }
}

// ---------------------------------------------------------------------------
// Kernel 4: masked mean pool over T.
// ---------------------------------------------------------------------------
__global__ __launch_bounds__(128) void pool_kernel(
    const float* __restrict__ z, const int* __restrict__ nbr_nids,
    float* __restrict__ z_link) {
  const int b = blockIdx.x, c = threadIdx.x;
  float s = 0.f;
  int cnt = 0;
  for (int t = 0; t < TT; t++) {
    bool valid = (nbr_nids[b * TT + t] != -1);
    if (valid) {
      s += z[(b * TT + t) * CC + c];
      cnt++;
    }
  }
  z_link[b * CC + c] = s / (float)(cnt > 0 ? cnt : 1);
}

// ---------------------------------------------------------------------------
// Kernel 5: deterministic CSR segment mean (binary search on sorted tg_seg)
// plus node embedding add.
// ---------------------------------------------------------------------------
__global__ __launch_bounds__(128) void seg_kernel(
    const float* __restrict__ node_feat, const int* __restrict__ tg_nbr_ids,
    const int* __restrict__ tg_seg, const int* __restrict__ node_ids,
    float* __restrict__ z_node) {
  const int b = blockIdx.x, c = threadIdx.x;
  int lo = 0, hi = EGN;
  while (lo < hi) {
    int m = (lo + hi) >> 1;
    if (tg_seg[m] < b) lo = m + 1; else hi = m;
  }
  int lo2 = lo, hi2 = EGN;
  while (lo2 < hi2) {
    int m = (lo2 + hi2) >> 1;
    if (tg_seg[m] < b + 1) lo2 = m + 1; else hi2 = m;
  }
  float s = 0.f;
  for (int e = lo; e < lo2; e++)
    s += node_feat[(size_t)tg_nbr_ids[e] * NDIM + c];
  int cnt = lo2 - lo;
  float agg = s / (float)(cnt > 0 ? cnt : 1);
  z_node[b * CC + c] = agg + node_feat[(size_t)node_ids[b] * NDIM + c];
}

// ---------------------------------------------------------------------------
// Kernel 6: out = concat(z_link, z_node) [B,256] @ out_w [256,128] + out_b
// ---------------------------------------------------------------------------
__global__ __launch_bounds__(256) void out_kernel(
    const float* __restrict__ z_link, const float* __restrict__ z_node,
    const float* __restrict__ out_w, const float* __restrict__ out_b,
    float* __restrict__ out) {
  __shared__ float xt[32 * 260]; // 32x256 padded
  const int tid = threadIdx.x;
  const int m0 = blockIdx.x * 32;
  for (int idx = tid; idx < 32 * 256; idx += 256) {
    int r = idx >> 8, c = idx & 255;
    float v = (c < CC) ? z_link[(m0 + r) * CC + c]
                       : z_node[(m0 + r) * CC + (c - CC)];
    xt[r * 260 + c] = v;
  }
  __syncthreads();

  const int lane = tid & 31, wave = tid >> 5;
  const int half = lane >> 4, l16 = lane & 15;
  const int n0 = wave * 16;
  v8f acc0, acc1;
  {
    float bias = out_b[n0 + l16];
#pragma unroll
    for (int i = 0; i < 8; i++) { acc0[i] = bias; acc1[i] = bias; }
  }
  for (int k0 = 0; k0 < 256; k0 += 4) {
    const int kk = k0 + 2 * half;
    v2f bb, a0, a1;
    bb.x = out_w[kk * CC + n0 + l16];
    bb.y = out_w[(kk + 1) * CC + n0 + l16];
    a0.x = xt[l16 * 260 + kk];
    a0.y = xt[l16 * 260 + kk + 1];
    a1.x = xt[(l16 + 16) * 260 + kk];
    a1.y = xt[(l16 + 16) * 260 + kk + 1];
    acc0 = wmma4(a0, bb, acc0);
    acc1 = wmma4(a1, bb, acc1);
  }
#pragma unroll
  for (int i = 0; i < 8; i++) {
    out[(m0 + i + 8 * half) * CC + n0 + l16] = acc0[i];
    out[(m0 + 16 + i + 8 * half) * CC + n0 + l16] = acc1[i];
  }
}

// ---------------------------------------------------------------------------
extern "C" void kernel_launch(void* const* d_in, const int* in_sizes, int n_in,
                              void* d_out, int out_size, void* d_ws,
                              size_t ws_size, hipStream_t stream) {
  const float* node_feat = (const float*)d_in[0];
  const float* nbr_feats = (const float*)d_in[1];
  const float* times = (const float*)d_in[2];
  const float* nbr_times = (const float*)d_in[3];
  const float* time_w = (const float*)d_in[4];
  const float* time_b = (const float*)d_in[5];
  const float* proj_w = (const float*)d_in[6];
  const float* proj_b = (const float*)d_in[7];
  const float* tok_gamma = (const float*)d_in[8];
  const float* tok_beta = (const float*)d_in[9];
  const float* tok_w1 = (const float*)d_in[10];
  const float* tok_b1 = (const float*)d_in[11];
  const float* tok_w2 = (const float*)d_in[12];
  const float* tok_b2 = (const float*)d_in[13];
  const float* ch_gamma = (const float*)d_in[14];
  const float* ch_beta = (const float*)d_in[15];
  const float* ch_w1 = (const float*)d_in[16];
  const float* ch_b1 = (const float*)d_in[17];
  const float* ch_w2 = (const float*)d_in[18];
  const float* ch_b2 = (const float*)d_in[19];
  const float* out_w = (const float*)d_in[20];
  const float* out_b = (const float*)d_in[21];
  const int* nbr_nids = (const int*)d_in[22];
  const int* tg_nbr_ids = (const int*)d_in[23];
  const int* tg_seg = (const int*)d_in[24];
  const int* node_ids = (const int*)d_in[25];
  float* out = (float*)d_out;

  float* z = (float*)d_ws;                  // [B*T, 128]
  float* z_link = z + (size_t)BB * TT * CC; // [B, 128]
  float* z_node = z_link + (size_t)BB * CC; // [B, 128]

  proj_kernel<<<(BB * TT) / 32, 256, 0, stream>>>(
      nbr_feats, times, nbr_times, time_w, time_b, proj_w, proj_b, z);
  for (int l = 0; l < 2; l++) {
    token_kernel<<<BB, 128, 0, stream>>>(tok_gamma, tok_beta, tok_w1, tok_b1,
                                         tok_w2, tok_b2, z, l);
    channel_kernel<<<(BB * TT) / 32, 256, 0, stream>>>(
        ch_gamma, ch_beta, ch_w1, ch_b1, ch_w2, ch_b2, z, l);
  }
  pool_kernel<<<BB, 128, 0, stream>>>(z, nbr_nids, z_link);
  seg_kernel<<<BB, 128, 0, stream>>>(node_feat, tg_nbr_ids, tg_seg, node_ids,
                                     z_node);
  out_kernel<<<BB / 32, 256, 0, stream>>>(z_link, z_node, out_w, out_b, out);
}